// GATv2_63943473103172
// MI455X (gfx1250) — compile-verified
//
#include <hip/hip_runtime.h>
#include <hip/hip_bf16.h>

typedef __attribute__((ext_vector_type(2))) float v2f;
typedef __attribute__((ext_vector_type(8))) float v8f;

#define FD 128      // feature dim
#define CD 128      // out channels
#define IND 256     // concat dim
#define NEG_SLOPE 0.2f

// ---- order-preserving float<->uint mapping for atomic max on floats ----
__device__ __forceinline__ unsigned f2o(float f) {
    unsigned u = __float_as_uint(f);
    return (u & 0x80000000u) ? ~u : (u | 0x80000000u);
}
__device__ __forceinline__ float o2f(unsigned u) {
    return (u & 0x80000000u) ? __uint_as_float(u & 0x7fffffffu)
                             : __uint_as_float(~u);
}
#define NEG_INF_ORD 0x007FFFFFu   // f2o(-inf)

// ---------------------------------------------------------------------------
// init: seed segment-max tables and denominators
// ---------------------------------------------------------------------------
__global__ void init_kernel(unsigned* __restrict__ agg_u,
                            unsigned* __restrict__ amax_u,
                            float* __restrict__ den, int N) {
    int i = blockIdx.x * 256 + threadIdx.x;
    if (i < N * CD) agg_u[i] = NEG_INF_ORD;
    if (i < N) { amax_u[i] = NEG_INF_ORD; den[i] = 0.0f; }
}

// ---------------------------------------------------------------------------
// node projections: xr = z@W_r, xl = z@W_l, XG = xl + graph_fts@W_e
//   z = concat(x + graph_fts[batch], hidden)   [16-row tiles per wave]
// ---------------------------------------------------------------------------
#define ZSTR 260   // padded LDS row stride (floats)

__global__ __launch_bounds__(128) void node_proj_kernel(
    const float* __restrict__ x, const float* __restrict__ graph_fts,
    const float* __restrict__ hidden, const int* __restrict__ batch,
    const float* __restrict__ Wl, const float* __restrict__ Wr,
    const float* __restrict__ We,
    float* __restrict__ xl, float* __restrict__ xr, float* __restrict__ XG,
    int N)
{
    extern __shared__ float lds[];
    const int lane = threadIdx.x & 31;
    const int wave = threadIdx.x >> 5;
    float* zt = lds + (size_t)wave * 16 * ZSTR;

    int tile = blockIdx.x * 4 + wave;
    int m0 = tile * 16;
    const bool active = (m0 < N);
    if (!active) m0 = 0;   // clamp: keep EXEC full for WMMA, skip stores

    const int hi = lane >> 4;      // half-wave select (K pairs)
    const int lm = lane & 15;      // row (A) / col (B,D) within tile

    // ---- stage z tile: 16 rows x 256 cols (float4 coalesced) ----
    for (int i = lane; i < 16 * 64; i += 32) {
        int r = i >> 6, c4 = (i & 63) * 4;
        int node = m0 + r;
        float4 v;
        if (c4 < FD) {
            float4 a = *(const float4*)(x + (size_t)node * FD + c4);
            int b = batch[node];
            float4 g = *(const float4*)(graph_fts + (size_t)b * FD + c4);
            v = make_float4(a.x + g.x, a.y + g.y, a.z + g.z, a.w + g.w);
        } else {
            v = *(const float4*)(hidden + (size_t)node * CD + (c4 - FD));
        }
        *(float4*)(zt + (size_t)r * ZSTR + c4) = v;
    }
    // per-wave LDS ops are in-order; tiles are wave-private: no barrier

    v8f acc[8];
    const v8f vzero = {0.f,0.f,0.f,0.f,0.f,0.f,0.f,0.f};

    auto zero_acc = [&]() {
        #pragma unroll
        for (int nt = 0; nt < 8; ++nt) acc[nt] = vzero;
    };
    auto run_gemm = [&](const float* __restrict__ W, int K) {
        for (int k0 = 0; k0 < K; k0 += 4) {
            const int kk = k0 + 2 * hi;
            v2f a = *(const v2f*)(zt + (size_t)lm * ZSTR + kk);
            #pragma unroll
            for (int nt = 0; nt < 8; ++nt) {
                v2f b;
                b.x = W[(size_t)kk * CD + nt * 16 + lm];
                b.y = W[(size_t)(kk + 1) * CD + nt * 16 + lm];
                acc[nt] = __builtin_amdgcn_wmma_f32_16x16x4_f32(
                    false, a, false, b, (short)0, acc[nt], false, false);
            }
        }
    };
    auto store_acc = [&](float* __restrict__ out) {
        if (!active) return;
        #pragma unroll
        for (int nt = 0; nt < 8; ++nt)
            #pragma unroll
            for (int v = 0; v < 8; ++v)
                out[(size_t)(m0 + v + 8 * hi) * CD + nt * 16 + lm] = acc[nt][v];
    };

    zero_acc(); run_gemm(Wr, IND); store_acc(xr);
    zero_acc(); run_gemm(Wl, IND); store_acc(xl);

    // ---- restage A = graph_fts rows; continue accumulating on xl acc ----
    for (int i = lane; i < 16 * 32; i += 32) {
        int r = i >> 5, c4 = (i & 31) * 4;
        *(float4*)(zt + (size_t)r * ZSTR + c4) =
            *(const float4*)(graph_fts + (size_t)(m0 + r) * FD + c4);
    }
    run_gemm(We, FD);       // acc = xl + graph_fts@W_e
    store_acc(XG);
}

// ---------------------------------------------------------------------------
// edge attention logits:
//   alpha[e] = leaky(edge_attr@W_e + XG[src] + xr[tgt]) . att ; amax = segmax
// W_e staged k-pair-interleaved so each B fetch is one conflict-free b64.
// ---------------------------------------------------------------------------
#define PSTR 144   // float2 row stride for paired W_e (halves bank-disjoint)
#define ASTR 132   // padded LDS stride for edge A tiles (conflict-free b64)

__global__ __launch_bounds__(256) void edge_alpha_kernel(
    const float* __restrict__ edge_attr, const float* __restrict__ We,
    const float* __restrict__ XG, const float* __restrict__ xr,
    const float* __restrict__ att,
    const int* __restrict__ src, const int* __restrict__ tgt,
    float* __restrict__ alpha, unsigned* __restrict__ amax_u, int E)
{
    extern __shared__ float lds[];
    float2* wp = (float2*)lds;                         // 64 x PSTR float2
    const int lane = threadIdx.x & 31;
    const int wave = threadIdx.x >> 5;
    float* at = lds + (size_t)64 * PSTR * 2 + (size_t)wave * 16 * ASTR;

    // ---- stage W_e as (k,k+1) pairs: wp[k2][col] = {W[2k2][col], W[2k2+1][col]}
    for (int i = threadIdx.x; i < 64 * 128; i += 256) {
        int k2 = i >> 7, col = i & 127;
        float2 v;
        v.x = We[(size_t)(2 * k2) * CD + col];
        v.y = We[(size_t)(2 * k2 + 1) * CD + col];
        wp[(size_t)k2 * PSTR + col] = v;
    }
    __syncthreads();

    int tile = blockIdx.x * 8 + wave;
    int e0 = tile * 16;
    const bool active = (e0 < E);
    if (!active) e0 = 0;

    const int hi = lane >> 4, lm = lane & 15;

    // ---- stage A tile: 16 consecutive edge rows (contiguous 8KB) ----
    for (int i = lane; i < 16 * 32; i += 32) {
        int r = i >> 5, c4 = (i & 31) * 4;
        *(float4*)(at + (size_t)r * ASTR + c4) =
            *(const float4*)(edge_attr + (size_t)(e0 + r) * FD + c4);
    }

    v8f acc[8];
    const v8f vzero = {0.f,0.f,0.f,0.f,0.f,0.f,0.f,0.f};
    #pragma unroll
    for (int nt = 0; nt < 8; ++nt) acc[nt] = vzero;

    for (int k0 = 0; k0 < FD; k0 += 4) {
        const int k2 = (k0 >> 1) + hi;                 // (k0 + 2*hi) / 2
        v2f a = *(const v2f*)(at + (size_t)lm * ASTR + k0 + 2 * hi);
        #pragma unroll
        for (int nt = 0; nt < 8; ++nt) {
            v2f b = *(const v2f*)(wp + (size_t)k2 * PSTR + nt * 16 + lm);
            acc[nt] = __builtin_amdgcn_wmma_f32_16x16x4_f32(
                false, a, false, b, (short)0, acc[nt], false, false);
        }
    }

    // ---- epilogue: gathers, leaky relu, dot with att, half-wave reduce ----
    float attv[8];
    #pragma unroll
    for (int nt = 0; nt < 8; ++nt) attv[nt] = att[nt * 16 + lm];

    int s_v[8], t_v[8];
    #pragma unroll
    for (int v = 0; v < 8; ++v) {
        int e = e0 + v + 8 * hi;
        s_v[v] = src[e];
        t_v[v] = tgt[e];
    }
    #pragma unroll
    for (int v = 0; v < 8; ++v) {
        float p = 0.0f;
        #pragma unroll
        for (int nt = 0; nt < 8; ++nt) {
            int col = nt * 16 + lm;
            float val = acc[nt][v]
                      + XG[(size_t)s_v[v] * CD + col]
                      + xr[(size_t)t_v[v] * CD + col];
            val = (val > 0.0f) ? val : NEG_SLOPE * val;
            p += val * attv[nt];
        }
        // reduce across the 16 lanes of this half-wave
        p += __shfl_xor(p, 1, 32);
        p += __shfl_xor(p, 2, 32);
        p += __shfl_xor(p, 4, 32);
        p += __shfl_xor(p, 8, 32);
        if (lm == 0 && active) {
            int e = e0 + v + 8 * hi;
            alpha[e] = p;
            atomicMax(amax_u + t_v[v], f2o(p));
        }
    }
}

// ---------------------------------------------------------------------------
// segment softmax numerator + denominator
// ---------------------------------------------------------------------------
__global__ void edge_softmax_kernel(const float* __restrict__ alpha,
                                    const unsigned* __restrict__ amax_u,
                                    const int* __restrict__ tgt,
                                    float* __restrict__ ex,
                                    float* __restrict__ den, int E) {
    int e = blockIdx.x * 256 + threadIdx.x;
    if (e >= E) return;
    int t = tgt[e];
    float am = o2f(amax_u[t]);
    if ((__float_as_uint(am) & 0x7F800000u) == 0x7F800000u) am = 0.0f; // inf/nan -> 0
    float ev = __expf(alpha[e] - am);
    ex[e] = ev;
    atomicAdd(den + t, ev);
}

// ---------------------------------------------------------------------------
// messages + scatter-max:  agg[t][c] = max over e: xl[src[e]][c] * a[e]
// one wave per edge, 4 channels per lane
// ---------------------------------------------------------------------------
__global__ void scatter_msg_kernel(const float* __restrict__ xl,
                                   const float* __restrict__ ex,
                                   const float* __restrict__ den,
                                   const int* __restrict__ src,
                                   const int* __restrict__ tgt,
                                   unsigned* __restrict__ agg_u, int E) {
    int gid = blockIdx.x * 256 + threadIdx.x;
    int e = gid >> 5;
    if (e >= E) return;
    int q = (gid & 31) * 4;
    int s = src[e], t = tgt[e];
    float d = fmaxf(den[t], 1e-16f);
    float a = ex[e] / d;
    float4 m = *(const float4*)(xl + (size_t)s * CD + q);
    unsigned* base = agg_u + (size_t)t * CD + q;
    atomicMax(base + 0, f2o(m.x * a));
    atomicMax(base + 1, f2o(m.y * a));
    atomicMax(base + 2, f2o(m.z * a));
    atomicMax(base + 3, f2o(m.w * a));
}

// ---------------------------------------------------------------------------
// finalize: decode, empty segments -> 0, add hidden
// ---------------------------------------------------------------------------
__global__ void finalize_kernel(const unsigned* __restrict__ agg_u,
                                const float* __restrict__ hidden,
                                float* __restrict__ out, int N) {
    int i = blockIdx.x * 256 + threadIdx.x;
    if (i >= N * CD) return;
    float f = o2f(agg_u[i]);
    if ((__float_as_uint(f) & 0x7F800000u) == 0x7F800000u) f = 0.0f; // -inf/nan -> 0
    out[i] = f + hidden[i];
}

// ---------------------------------------------------------------------------
extern "C" void kernel_launch(void* const* d_in, const int* in_sizes, int n_in,
                              void* d_out, int out_size, void* d_ws, size_t ws_size,
                              hipStream_t stream) {
    const float* x          = (const float*)d_in[0];
    const float* edge_attr  = (const float*)d_in[1];
    const float* graph_fts  = (const float*)d_in[2];
    const float* hidden     = (const float*)d_in[3];
    const float* edges_hid  = (const float*)d_in[4];
    const float* W_l        = (const float*)d_in[5];
    const float* W_r        = (const float*)d_in[6];
    const float* W_e        = (const float*)d_in[7];
    const float* att        = (const float*)d_in[8];
    const int*   edge_index = (const int*)d_in[9];
    const int*   batch      = (const int*)d_in[10];

    const int N = in_sizes[0] / FD;       // 50000
    const int E = in_sizes[1] / FD;       // 800000
    const int* src = edge_index;
    const int* tgt = edge_index + E;

    // workspace layout (f32 / u32 words)
    float* xl    = (float*)d_ws;              // N*128
    float* xr    = xl + (size_t)N * CD;       // N*128
    float* XG    = xr + (size_t)N * CD;       // N*128 (= xl + graph_fts@W_e)
    float* alpha = XG + (size_t)N * CD;       // E
    float* exb   = alpha + (size_t)E;         // E
    float* den   = exb + (size_t)E;           // N
    unsigned* amax_u = (unsigned*)(den + N);  // N
    unsigned* agg_u  = amax_u + N;            // N*128

    float* out0 = (float*)d_out;              // [N, C]
    float* out1 = out0 + (size_t)N * CD;      // [E, F] passthrough

    // 1) init segment tables
    {
        int total = N * CD;
        init_kernel<<<(total + 255) / 256, 256, 0, stream>>>(agg_u, amax_u, den, N);
    }
    // 2) node projections via WMMA
    {
        int tiles = (N + 15) / 16;
        int blocks = (tiles + 3) / 4;
        size_t smem = (size_t)4 * 16 * ZSTR * sizeof(float);   // ~65 KB
        node_proj_kernel<<<blocks, 128, smem, stream>>>(
            x, graph_fts, hidden, batch, W_l, W_r, W_e, xl, xr, XG, N);
    }
    // 3) edge attention logits via WMMA + segment max
    {
        int tiles = (E + 15) / 16;
        int blocks = (tiles + 7) / 8;
        size_t smem = (size_t)64 * PSTR * 2 * sizeof(float)
                    + (size_t)8 * 16 * ASTR * sizeof(float);   // ~138 KB
        edge_alpha_kernel<<<blocks, 256, smem, stream>>>(
            edge_attr, W_e, XG, xr, att, src, tgt, alpha, amax_u, E);
    }
    // 4) softmax numerators + denominators
    edge_softmax_kernel<<<(E + 255) / 256, 256, 0, stream>>>(alpha, amax_u, tgt, exb, den, E);
    // 5) messages + scatter max (one wave per edge)
    {
        long threads = (long)E * 32;
        scatter_msg_kernel<<<(int)((threads + 255) / 256), 256, 0, stream>>>(
            xl, exb, den, src, tgt, agg_u, E);
    }
    // 6) finalize node output
    finalize_kernel<<<(N * CD + 255) / 256, 256, 0, stream>>>(agg_u, hidden, out0, N);
    // 7) edges_hidden passthrough
    hipMemcpyAsync(out1, edges_hid, (size_t)E * FD * sizeof(float),
                   hipMemcpyDeviceToDevice, stream);
}